// OperatorModule_11166914970006
// MI455X (gfx1250) — compile-verified
//
#include <hip/hip_runtime.h>
#include <hip/hip_bf16.h>
#include <stdint.h>

#define IMG_H 512
#define IMG_W 512
#define NDET  512
#define NSAMP 512
#define NANG  360
#define BC    32            // B*C = 4*8 channels
#define HW    (IMG_H * IMG_W)

// ---------------------------------------------------------------------------
// Pass 1: transpose img (BC, H, W) -> imgT (H, W, BC) so that the 32 channel
// values of any pixel are one contiguous 128B line (one line per wave32 tap).
// Uses gfx1250 async global->LDS loads (ASYNCcnt) to stage 32 channel rows,
// then writes the interleaved layout back out fully coalesced.
// ---------------------------------------------------------------------------
#define TP_XS 256

__global__ __launch_bounds__(256)
void radon_transpose_kernel(const float* __restrict__ img,
                            float* __restrict__ imgT) {
  __shared__ float tile[BC][TP_XS + 1];          // +1 pad: no bank conflicts
  const int t  = threadIdx.x;                    // 0..255  (x within strip)
  const int y  = blockIdx.y;
  const int x0 = blockIdx.x * TP_XS;

  // Wave-uniform 64-bit base; per-lane 32-bit offset (GVS addressing).
  const float* gbase = img + (size_t)y * IMG_W + x0;
  const unsigned lds_base = (unsigned)(uintptr_t)&tile[0][0];

#pragma unroll
  for (int ch = 0; ch < BC; ++ch) {
    unsigned lds_off = lds_base + (unsigned)((ch * (TP_XS + 1) + t) * 4);
    unsigned voff    = (unsigned)((ch * HW + t) * 4);
    asm volatile("global_load_async_to_lds_b32 %0, %1, %2"
                 :
                 : "v"(lds_off), "v"(voff), "s"(gbase)
                 : "memory");
  }
  asm volatile("s_wait_asynccnt 0" ::: "memory");  // this wave's LDS writes done
  __syncthreads();                                 // all waves' LDS writes done

  // Coalesced write-out: flat j*256+t maps to (x_loc = flat>>5, ch = flat&31)
  float* obase = imgT + ((size_t)y * IMG_W + x0) * BC;
#pragma unroll
  for (int j = 0; j < BC; ++j) {
    int flat  = j * 256 + t;
    int x_loc = flat >> 5;
    int ch    = flat & 31;
    obase[flat] = tile[ch][x_loc];
  }
}

// ---------------------------------------------------------------------------
// Pass 2: forward projection. One wave per (angle, detector); lane = channel.
// All address/weight math is wave-uniform (amortized over 32 channels);
// each of the 4 bilinear taps is a single 128B coalesced line in imgT layout.
// Sample loop is clamped to the analytic ray/volume intersection (uniform
// branch, wave32), with the reference's exact per-sample validity test kept.
// ---------------------------------------------------------------------------
template <bool TRANSPOSED>
__global__ __launch_bounds__(256)
void radon_main_kernel(const float* __restrict__ src,
                       const float* __restrict__ angles,
                       float* __restrict__ out) {
  const int lane = threadIdx.x & 31;
  const int wave = threadIdx.x >> 5;
  const int a    = blockIdx.y;
  const int d    = blockIdx.x * 8 + wave;

  float sn, c;
  sincosf(angles[a], &sn, &c);

  const float tdet = (float)d - 255.5f;
  const float Ax   = 255.5f + tdet * c;    // xs = Ax - s*sn
  const float By   = 255.5f + tdet * sn;   // ys = By + s*c

  // Analytic s-interval where the ray can be inside the volume (with margin;
  // the exact reference `valid` test below handles the boundary precisely).
  float lo = -256.5f, hi = 256.5f;
  if (sn > 1e-6f)       { lo = fmaxf(lo, (Ax - 511.0f) / sn); hi = fminf(hi, Ax / sn); }
  else if (sn < -1e-6f) { lo = fmaxf(lo, Ax / sn);            hi = fminf(hi, (Ax - 511.0f) / sn); }
  else if (Ax < -1.0f || Ax > 512.0f) { lo = 1.0f; hi = 0.0f; }
  if (c > 1e-6f)        { lo = fmaxf(lo, -By / c);            hi = fminf(hi, (511.0f - By) / c); }
  else if (c < -1e-6f)  { lo = fmaxf(lo, (511.0f - By) / c);  hi = fminf(hi, -By / c); }
  else if (By < -1.0f || By > 512.0f) { lo = 1.0f; hi = 0.0f; }

  int silo = max(0,         (int)ceilf(lo + 255.5f) - 1);
  int sihi = min(NSAMP - 1, (int)floorf(hi + 255.5f) + 1);

  float acc = 0.0f;
#pragma unroll 2
  for (int si = silo; si <= sihi; ++si) {
    const float s   = (float)si - 255.5f;
    const float xs  = fmaf(-sn, s, Ax);
    const float ys  = fmaf(c, s, By);
    const float x0f = floorf(xs);
    const float y0f = floorf(ys);
    // reference: valid = x0>=0 & x0+1<=511 & y0>=0 & y0+1<=511
    if (x0f >= 0.0f && x0f <= 510.0f && y0f >= 0.0f && y0f <= 510.0f) {
      const float wx = xs - x0f;
      const float wy = ys - y0f;
      const int   xi = (int)x0f;
      const int   yi = (int)y0f;
      float v00, v01, v10, v11;
      if (TRANSPOSED) {
        const float* p = src + (size_t)(yi * IMG_W + xi) * BC + lane;
        v00 = p[0];             v01 = p[BC];
        v10 = p[IMG_W * BC];    v11 = p[IMG_W * BC + BC];
      } else {
        const float* p = src + (size_t)lane * HW + yi * IMG_W + xi;
        v00 = p[0];             v01 = p[1];
        v10 = p[IMG_W];         v11 = p[IMG_W + 1];
      }
      const float top = fmaf(wx, v01 - v00, v00);
      const float bot = fmaf(wx, v11 - v10, v10);
      acc += fmaf(wy, bot - top, top);
    }
  }

  // out layout (B, C, A, D): bc-major, lane == bc
  out[(size_t)lane * (NANG * NDET) + (size_t)a * NDET + d] = acc;
}

// ---------------------------------------------------------------------------
extern "C" void kernel_launch(void* const* d_in, const int* in_sizes, int n_in,
                              void* d_out, int out_size, void* d_ws, size_t ws_size,
                              hipStream_t stream) {
  const float* x      = (const float*)d_in[0];   // (4, 8, 512, 512) fp32
  const float* angles = (const float*)d_in[1];   // (360,) fp32
  float*       out    = (float*)d_out;           // (4, 8, 360, 512) fp32

  const size_t needT = (size_t)HW * BC * sizeof(float);  // 32 MB transposed copy
  if (ws_size >= needT) {
    float* imgT = (float*)d_ws;
    radon_transpose_kernel<<<dim3(IMG_W / TP_XS, IMG_H), 256, 0, stream>>>(x, imgT);
    radon_main_kernel<true><<<dim3(NDET / 8, NANG), 256, 0, stream>>>(imgT, angles, out);
  } else {
    radon_main_kernel<false><<<dim3(NDET / 8, NANG), 256, 0, stream>>>(x, angles, out);
  }
}